// GNN_Variant_47914655154257
// MI455X (gfx1250) — compile-verified
//
#include <hip/hip_runtime.h>

// ---------------------------------------------------------------------------
// GIN forward for MI455X (gfx1250, wave32).
// Aggregation (scatter-add) is the bound; GEMMs use native fp32 WMMA
// (V_WMMA_F32_16X16X4_F32) so precision matches the fp32 reference.
// ---------------------------------------------------------------------------

typedef float v2f __attribute__((ext_vector_type(2)));
typedef float v8f __attribute__((ext_vector_type(8)));

#define DIMF 128
#define LDS_STRIDE 132   // pad 128 -> 132 floats: conflict-free b64 LDS reads

__global__ void zero_kernel(float* p, int n) {
    int i = blockIdx.x * blockDim.x + threadIdx.x;
    if (i < n) p[i] = 0.0f;
}

__global__ void copy_kernel(float* __restrict__ dst, const float* __restrict__ src, size_t n4) {
    size_t i = (size_t)blockIdx.x * blockDim.x + threadIdx.x;
    if (i < n4) ((float4*)dst)[i] = ((const float4*)src)[i];
}

// agg[row[e]] += h[col[e]] for row!=col; one thread per (edge, 4-col chunk)
__global__ void scatter_edges(const float* __restrict__ h,
                              const int* __restrict__ row,
                              const int* __restrict__ col,
                              float* __restrict__ agg, int nE) {
    long long tid = (long long)blockIdx.x * blockDim.x + threadIdx.x;
    int e  = (int)(tid >> 5);
    int c4 = (int)(tid & 31);
    if (e >= nE) return;
    int r = row[e], c = col[e];
    if (r == c) return;                           // remove_self_loops
    float4 v = ((const float4*)(h + (size_t)c * DIMF))[c4];
    float* dst = agg + (size_t)r * DIMF + (size_t)c4 * 4;
    unsafeAtomicAdd(dst + 0, v.x);
    unsafeAtomicAdd(dst + 1, v.y);
    unsafeAtomicAdd(dst + 2, v.z);
    unsafeAtomicAdd(dst + 3, v.w);
}

// z[batch[n], colOff..colOff+127] += h[n, :]   (global_add_pool)
__global__ void readout_kernel(const float* __restrict__ h,
                               const int* __restrict__ batch,
                               float* __restrict__ z, int colOff, int nN) {
    long long tid = (long long)blockIdx.x * blockDim.x + threadIdx.x;
    int n  = (int)(tid >> 5);
    int c4 = (int)(tid & 31);
    if (n >= nN) return;
    int g = batch[n];
    float4 v = ((const float4*)(h + (size_t)n * DIMF))[c4];
    float* dst = z + (size_t)g * 640 + colOff + (size_t)c4 * 4;
    unsafeAtomicAdd(dst + 0, v.x);
    unsafeAtomicAdd(dst + 1, v.y);
    unsafeAtomicAdd(dst + 2, v.z);
    unsafeAtomicAdd(dst + 3, v.w);
}

// out = BN(ReLU(in @ W + b)) for one GIN MLP sublayer.  in: [nRows x 128],
// W: [128 x 128] row-major.  8 waves/block, one 16-row tile per wave,
// full N=128 strip (8 WMMA n-tiles) per wave, fp32 WMMA 16x16x4, K-loop of 32.
__global__ __launch_bounds__(256)
void gin_mlp_gemm(const float* __restrict__ in, const float* __restrict__ W,
                  const float* __restrict__ bias,
                  const float* __restrict__ gamma, const float* __restrict__ beta,
                  const float* __restrict__ mean,  const float* __restrict__ var,
                  float* __restrict__ out, int nRows) {
    __shared__ float sWt[DIMF * LDS_STRIDE];       // W transposed: sWt[col*132 + row]
    __shared__ float sA[8 * 16 * LDS_STRIDE];      // per-wave 16x128 A tile

    const int tid  = threadIdx.x;
    const int lane = tid & 31;
    const int wave = tid >> 5;

    // cooperative transpose-load of W into LDS (coalesced global reads)
    for (int i = tid; i < DIMF * DIMF; i += 256) {
        int r = i >> 7;      // row of W
        int c = i & 127;     // col of W
        sWt[c * LDS_STRIDE + r] = W[i];
    }
    __syncthreads();

    const int tile     = blockIdx.x * 8 + wave;
    const int numTiles = nRows >> 4;
    if (tile >= numTiles) return;                  // wave-uniform; EXEC stays full

    // stage this wave's 16x128 A tile into LDS (rows are 512B, float4 coalesced)
    float* sAw = sA + wave * (16 * LDS_STRIDE);
    const float* gA = in + (size_t)tile * 16 * DIMF;
    for (int i = lane; i < 16 * 32; i += 32) {
        int r  = i >> 5;
        int c4 = i & 31;
        float4 v = ((const float4*)(gA + r * DIMF))[c4];
        float* d = sAw + r * LDS_STRIDE + c4 * 4;
        d[0] = v.x; d[1] = v.y; d[2] = v.z; d[3] = v.w;
    }

    v8f acc[8] = {};

    // fp32 WMMA operand layout (ISA 7.12.2):
    //   A 16x4:  lanes 0-15 -> M=lane, K = k+{0,1};  lanes 16-31 -> K = k+{2,3}
    //   B 4x16:  lanes 0-15 -> N=lane, K = k+{0,1};  lanes 16-31 -> K = k+{2,3}
    const int half = lane >> 4;          // 0 / 1
    const int l16  = lane & 15;
    const float* aRow = sAw + l16 * LDS_STRIDE + 2 * half;
    const float* bCol = sWt + l16 * LDS_STRIDE + 2 * half;

    for (int k = 0; k < DIMF; k += 4) {
        v2f a; a.x = aRow[k]; a.y = aRow[k + 1];
#pragma unroll
        for (int t = 0; t < 8; ++t) {
            const float* bp = bCol + t * 16 * LDS_STRIDE + k;
            v2f b; b.x = bp[0]; b.y = bp[1];
            acc[t] = __builtin_amdgcn_wmma_f32_16x16x4_f32(
                false, a, false, b, (short)0, acc[t], false, false);
        }
    }

    // epilogue: +bias, ReLU, BatchNorm(eval), store.
    // C/D layout: VGPR v -> M = v + 8*half, N = t*16 + l16
#pragma unroll
    for (int t = 0; t < 8; ++t) {
        int colg = t * 16 + l16;
        float bv = bias[colg];
        float sc = gamma[colg] * rsqrtf(var[colg] + 1e-5f);
        float sh = beta[colg] - mean[colg] * sc;
#pragma unroll
        for (int v = 0; v < 8; ++v) {
            int m = v + 8 * half;
            float xv = acc[t][v] + bv;
            xv = fmaxf(xv, 0.0f);
            xv = xv * sc + sh;
            out[((size_t)tile * 16 + m) * DIMF + colg] = xv;
        }
    }
}

// t1 = relu(z @ Wc1 + bc1), z: [64 x 640], Wc1: [640 x 640]
__global__ void fc1_kernel(const float* __restrict__ z, const float* __restrict__ Wc1,
                           const float* __restrict__ bc1, float* __restrict__ t1) {
    int i = blockIdx.x * blockDim.x + threadIdx.x;
    if (i >= 64 * 640) return;
    int g = i / 640, c = i % 640;
    const float* zr = z + (size_t)g * 640;
    float s = bc1[c];
    for (int k = 0; k < 640; ++k) s = fmaf(zr[k], Wc1[(size_t)k * 640 + c], s);
    t1[i] = fmaxf(s, 0.0f);
}

// out = log_softmax(t1 @ Wc2 + bc2); one wave per graph
__global__ void fc2_lsm_kernel(const float* __restrict__ t1, const float* __restrict__ Wc2,
                               const float* __restrict__ bc2, float* __restrict__ out) {
    int g = blockIdx.x;
    int t = threadIdx.x;
    __shared__ float logits[10];
    if (t < 10) {
        const float* tr = t1 + (size_t)g * 640;
        float s = bc2[t];
        for (int k = 0; k < 640; ++k) s = fmaf(tr[k], Wc2[(size_t)k * 10 + t], s);
        logits[t] = s;
    }
    __syncthreads();
    if (t < 10) {
        float mx = logits[0];
        for (int i = 1; i < 10; ++i) mx = fmaxf(mx, logits[i]);
        float se = 0.0f;
        for (int i = 0; i < 10; ++i) se += expf(logits[i] - mx);
        out[g * 10 + t] = logits[t] - mx - logf(se);
    }
}

extern "C" void kernel_launch(void* const* d_in, const int* in_sizes, int n_in,
                              void* d_out, int out_size, void* d_ws, size_t ws_size,
                              hipStream_t stream) {
    const float* x     = (const float*)d_in[0];
    const int*   ei    = (const int*)  d_in[1];
    const int*   batch = (const int*)  d_in[2];
    const float* W_mlp = (const float*)d_in[3];
    const float* b_mlp = (const float*)d_in[4];
    const float* bn_g  = (const float*)d_in[5];
    const float* bn_b  = (const float*)d_in[6];
    const float* bn_m  = (const float*)d_in[7];
    const float* bn_v  = (const float*)d_in[8];
    const float* Wc1   = (const float*)d_in[9];
    const float* bc1   = (const float*)d_in[10];
    const float* Wc2   = (const float*)d_in[11];
    const float* bc2   = (const float*)d_in[12];

    const int Nn = 50000, Ee = 600000, Ll = 4, Mm = 2;
    const size_t NB = (size_t)Nn * DIMF;           // floats per node matrix

    float* agg  = (float*)d_ws;
    float* tmp  = agg  + NB;
    float* hbuf = tmp  + NB;
    float* z    = hbuf + NB;                       // [64 x 640]
    float* t1   = z    + 64 * 640;                 // [64 x 640]

    const int* row = ei;
    const int* col = ei + Ee;

    zero_kernel<<<(64 * 640 + 255) / 256, 256, 0, stream>>>(z, 64 * 640);
    readout_kernel<<<(int)(((long long)Nn * 32 + 255) / 256), 256, 0, stream>>>(
        x, batch, z, 0, Nn);

    const float* hin = x;
    const int gemmBlocks = ((Nn >> 4) + 7) / 8;    // 3125 tiles / 8 waves

    for (int k = 0; k < Ll; ++k) {
        // agg = hin  (self-loop term), then scatter-add neighbors
        copy_kernel<<<(int)((NB / 4 + 255) / 256), 256, 0, stream>>>(agg, hin, NB / 4);
        scatter_edges<<<(int)(((long long)Ee * 32 + 255) / 256), 256, 0, stream>>>(
            hin, row, col, agg, Ee);

        const float* W0 = W_mlp + (size_t)(k * Mm + 0) * DIMF * DIMF;
        const float* W1 = W_mlp + (size_t)(k * Mm + 1) * DIMF * DIMF;
        const size_t p0 = (size_t)(k * Mm + 0) * DIMF;
        const size_t p1 = (size_t)(k * Mm + 1) * DIMF;

        gin_mlp_gemm<<<gemmBlocks, 256, 0, stream>>>(
            agg, W0, b_mlp + p0, bn_g + p0, bn_b + p0, bn_m + p0, bn_v + p0, tmp, Nn);
        gin_mlp_gemm<<<gemmBlocks, 256, 0, stream>>>(
            tmp, W1, b_mlp + p1, bn_g + p1, bn_b + p1, bn_m + p1, bn_v + p1, hbuf, Nn);

        readout_kernel<<<(int)(((long long)Nn * 32 + 255) / 256), 256, 0, stream>>>(
            hbuf, batch, z, DIMF * (k + 1), Nn);
        hin = hbuf;
    }

    fc1_kernel<<<(64 * 640 + 255) / 256, 256, 0, stream>>>(z, Wc1, bc1, t1);
    fc2_lsm_kernel<<<64, 32, 0, stream>>>(t1, Wc2, bc2, (float*)d_out);
}